// PNA_81157702025502
// MI455X (gfx1250) — compile-verified
//
#include <hip/hip_runtime.h>

#define NN      100000
#define NE      300000
#define FD      64
#define DEF     4
#define NL      4
#define DELTA_C 2.5f
#define EPS_C   1e-5f
#define NF_TOT  (NN * FD)

typedef __attribute__((ext_vector_type(2))) float    v2f;
typedef __attribute__((ext_vector_type(8))) float    v8f;
typedef __attribute__((ext_vector_type(4))) unsigned v4u;
typedef __attribute__((ext_vector_type(8))) int      v8i;
typedef __attribute__((ext_vector_type(4))) int      v4i;

// CDNA5 f32 WMMA: D(16x16) = A(16x4) * B(4x16) + C
__device__ __forceinline__ v8f wmma4(v2f a, v2f b, v8f c) {
  return __builtin_amdgcn_wmma_f32_16x16x4_f32(false, a, false, b, (short)0, c,
                                               false, false);
}

// order-preserving float<->uint encoding for atomicMax/Min on floats
__device__ __forceinline__ unsigned fenc(float f) {
  unsigned u = __float_as_uint(f);
  return (u & 0x80000000u) ? ~u : (u | 0x80000000u);
}
__device__ __forceinline__ float fdec(unsigned u) {
  unsigned v = (u & 0x80000000u) ? (u ^ 0x80000000u) : ~u;
  return __uint_as_float(v);
}
#define ENC_NEGINF 0x007FFFFFu /* fenc(-inf) */
#define ENC_POSINF 0xFF800000u /* fenc(+inf) */

// Issue one TDM load of a 2-D f32 tensor (rows x 64, stride 64) into LDS.
// D# packing per CDNA5 ISA ch.8: group0 = {count/type/addr}, group1 = dims.
// This toolchain exposes the 6-arg builtin (extra zero-filled group).
__device__ __forceinline__ void tdm_load_2d_f32(unsigned lds_addr,
                                                const float* gptr, int rows) {
  unsigned long long ga = (unsigned long long)gptr;
  v4u g0;
  g0.x = 1u;                                       // count=1, user descriptor
  g0.y = lds_addr;                                 // lds_addr (bytes)
  g0.z = (unsigned)ga;                             // global_addr[31:0]
  g0.w = ((unsigned)(ga >> 32) & 0x01FFFFFFu) | 0x80000000u;  // addr[56:32]|type=2
  v8i g1;
  g1[0] = (int)(2u << 16);          // workgroup_mask=0, data_size=4B
  g1[1] = (int)(64u << 16);         // tensor_dim0=64 (low 16 @bits63:48)
  g1[2] = (int)((unsigned)rows << 16);  // dim0 hi=0 | tensor_dim1 low16
  g1[3] = (int)(64u << 16);         // dim1 hi=0 | tile_dim0=64
  g1[4] = rows;                     // tile_dim1=rows, tile_dim2=0
  g1[5] = 64;                       // tensor_dim0_stride=64 (low 32)
  g1[6] = 0;                        // stride0 hi | tensor_dim1_stride low
  g1[7] = 0;
  v4i gz4 = {0, 0, 0, 0};           // groups 2/3: unused (2-D tensor)
  v8i gz8 = {0, 0, 0, 0, 0, 0, 0, 0};
  __builtin_amdgcn_tensor_load_to_lds(g0, g1, gz4, gz4, gz8, 0);
}

// ---------------------------------------------------------------- init
__global__ void k_init(const float* __restrict__ nf, float* __restrict__ x,
                       float* __restrict__ deg) {
  int i = blockIdx.x * blockDim.x + threadIdx.x;
  if (i < NF_TOT) x[i] = nf[i];
  if (i < NN) deg[i] = 0.f;
}

__global__ void k_deg(const int* __restrict__ ei, float* __restrict__ deg) {
  int e = blockIdx.x * blockDim.x + threadIdx.x;
  if (e < NE) atomicAdd(deg + ei[NE + e], 1.f);
}

__global__ void k_reset(float* __restrict__ ssum, float* __restrict__ ssq,
                        unsigned* __restrict__ smax,
                        unsigned* __restrict__ smin,
                        float* __restrict__ bnstats) {
  int i = blockIdx.x * blockDim.x + threadIdx.x;
  if (i < NF_TOT) {
    ssum[i] = 0.f;
    ssq[i]  = 0.f;
    smax[i] = ENC_NEGINF;
    smin[i] = ENC_POSINF;
  }
  if (i < 2 * FD) bnstats[i] = 0.f;
}

// ---------------------------------------------------------------- edge MLP + scatter
// wave = 16 edges x 64 outputs; m = h @ Wpre + b,  h = [x_dst | x_src | e].
// Wpre (192x64, 48KB) staged once per block into LDS by the Tensor Data Mover;
// B-fragments then come from LDS, keeping the VMEM pipe free for gathers+atomics.
__global__ __launch_bounds__(256) void k_edge(
    const float* __restrict__ x, const int* __restrict__ ei,
    const float* __restrict__ ef, const float* __restrict__ Wew,
    const float* __restrict__ Web, const float* __restrict__ Wpw,
    const float* __restrict__ Wpb, float* __restrict__ ssum,
    float* __restrict__ ssq, unsigned* __restrict__ smax,
    unsigned* __restrict__ smin) {
  __shared__ float sW[192 * FD];  // 48 KB staged weights

  if (threadIdx.x < 32) {  // wave 0 drives the TDM; EXEC ignored by tensor op
    tdm_load_2d_f32((unsigned)(size_t)&sW[0], Wpw, 192);
    __builtin_amdgcn_s_wait_tensorcnt(0);
  }
  __syncthreads();  // all waves: weights now resident in LDS

  const int lane = threadIdx.x & 31;
  const int wv   = threadIdx.x >> 5;
  const int tile = blockIdx.x * 8 + wv;
  if (tile * 16 >= NE) return;  // after the barrier: safe
  const int m  = lane & 15;
  const int kh = (lane < 16) ? 0 : 2;
  const int em = tile * 16 + m;
  const int srcm = ei[em];
  const int dstm = ei[NE + em];
  const float ef0 = ef[em * 4 + 0], ef1 = ef[em * 4 + 1];
  const float ef2 = ef[em * 4 + 2], ef3 = ef[em * 4 + 3];

  const v8f vz = {0.f, 0.f, 0.f, 0.f, 0.f, 0.f, 0.f, 0.f};
  v8f acc[4] = {vz, vz, vz, vz};

  // part 1: rows 0..63 of Wpre multiply x[dst]
  const float* xd = x + (size_t)dstm * FD;
  for (int k0 = 0; k0 < FD; k0 += 4) {
    const int k = k0 + kh;
    v2f a = {xd[k], xd[k + 1]};
#pragma unroll
    for (int nt = 0; nt < 4; ++nt) {
      const float* wp = sW + k * FD + nt * 16 + m;
      v2f b = {wp[0], wp[FD]};
      acc[nt] = wmma4(a, b, acc[nt]);
    }
  }
  // part 2: rows 64..127 multiply x[src]
  const float* xs = x + (size_t)srcm * FD;
  for (int k0 = 0; k0 < FD; k0 += 4) {
    const int k = k0 + kh;
    v2f a = {xs[k], xs[k + 1]};
#pragma unroll
    for (int nt = 0; nt < 4; ++nt) {
      const float* wp = sW + (FD + k) * FD + nt * 16 + m;
      v2f b = {wp[0], wp[FD]};
      acc[nt] = wmma4(a, b, acc[nt]);
    }
  }
  // part 3: rows 128..191 multiply e = ef @ We + be (computed inline)
  for (int k0 = 0; k0 < FD; k0 += 4) {
    const int k = k0 + kh;
    float e0 = Web[k] + ef0 * Wew[k] + ef1 * Wew[FD + k] +
               ef2 * Wew[2 * FD + k] + ef3 * Wew[3 * FD + k];
    float e1 = Web[k + 1] + ef0 * Wew[k + 1] + ef1 * Wew[FD + k + 1] +
               ef2 * Wew[2 * FD + k + 1] + ef3 * Wew[3 * FD + k + 1];
    v2f a = {e0, e1};
#pragma unroll
    for (int nt = 0; nt < 4; ++nt) {
      const float* wp = sW + (2 * FD + k) * FD + nt * 16 + m;
      v2f b = {wp[0], wp[FD]};
      acc[nt] = wmma4(a, b, acc[nt]);
    }
  }

  // epilogue: bias + 4 segment reductions into L2-resident accumulators
#pragma unroll
  for (int v = 0; v < 8; ++v) {
    const int er = tile * 16 + v + ((lane < 16) ? 0 : 8);
    const int dn = ei[NE + er];
    const size_t base = (size_t)dn * FD;
#pragma unroll
    for (int nt = 0; nt < 4; ++nt) {
      const int col = nt * 16 + m;
      const float val = acc[nt][v] + Wpb[col];
      atomicAdd(ssum + base + col, val);
      atomicAdd(ssq + base + col, val * val);
      atomicMax(smax + base + col, fenc(val));
      atomicMin(smin + base + col, fenc(val));
    }
  }
}

// ---------------------------------------------------------------- node post-MLP
// wave = 16 nodes x 64 outputs; 2 waves/block (6250 tiles = 2*3125, no tails).
// out = x@W[0:64] + a6@W[64:448] + amp*(a6@W[448:832]) + att*(a6@W[832:1216]) + b
// Wlin staged into LDS with async global->LDS loads, overlapped with the big GEMM.
__global__ __launch_bounds__(64) void k_node(
    const float* __restrict__ x, const float* __restrict__ ssum,
    const float* __restrict__ ssq, const unsigned* __restrict__ smax,
    const unsigned* __restrict__ smin, const float* __restrict__ deg,
    const float* __restrict__ Wpow, const float* __restrict__ Wpob,
    const float* __restrict__ Wlw, const float* __restrict__ Wlb,
    float* __restrict__ out, float* __restrict__ bnsum,
    float* __restrict__ bnsq) {
  __shared__ float sWl[FD * FD];      // 16 KB: Wlin, filled asynchronously
  __shared__ float sOut[2][16 * FD];  // 8 KB: per-wave intermediate tile

  // kick off async LDS fill of Wlin (ASYNCcnt-tracked), consumed before GEMM2
  {
    const unsigned ldsb = (unsigned)(size_t)&sWl[0];
    const unsigned long long gb = (unsigned long long)Wlw;
    const unsigned t = threadIdx.x;  // 64 threads x 16 x 16B = 16 KB
#pragma unroll
    for (int r = 0; r < 16; ++r) {
      const unsigned off = (r * 64u + t) * 16u;
      unsigned la = ldsb + off;
      unsigned long long ga = gb + off;
      asm volatile("global_load_async_to_lds_b128 %0, %1, off" ::"v"(la),
                   "v"(ga)
                   : "memory");
    }
  }

  const int lane = threadIdx.x & 31;
  const int wv   = threadIdx.x >> 5;
  const int tile = blockIdx.x * 2 + wv;  // always < 6250
  const int m  = lane & 15;
  const int kh = (lane < 16) ? 0 : 2;
  const int nodem = tile * 16 + m;

  const float dv   = deg[nodem];
  const float degc = fmaxf(dv, 1.f);
  const float rdeg = 1.f / degc;
  const float logd = logf(degc + 1.f);
  const float amp  = logd * (1.f / DELTA_C);
  const float att  = DELTA_C / logd;
  const float hasn = (dv > 0.f) ? 1.f : 0.f;

  const float* xr = x + (size_t)nodem * FD;
  const float* sb = ssum + (size_t)nodem * FD;
  const float* qb = ssq + (size_t)nodem * FD;
  const unsigned* mxb = smax + (size_t)nodem * FD;
  const unsigned* mnb = smin + (size_t)nodem * FD;

  const v8f vz = {0.f, 0.f, 0.f, 0.f, 0.f, 0.f, 0.f, 0.f};
  v8f acc[4] = {vz, vz, vz, vz};

  // part 1: x part (Wpost rows 0..63)
  for (int k0 = 0; k0 < FD; k0 += 4) {
    const int k = k0 + kh;
    v2f a = {xr[k], xr[k + 1]};
#pragma unroll
    for (int nt = 0; nt < 4; ++nt) {
      const float* wp = Wpow + (size_t)k * FD + nt * 16 + m;
      v2f b = {wp[0], wp[FD]};
      acc[nt] = wmma4(a, b, acc[nt]);
    }
  }

  // part 2: six aggregators, each K=64, with amp/att variants (A-frag scaling)
  for (int t = 0; t < 16; ++t) {
    const int j = t * 4 + kh;
    const float s0 = sb[j], s1 = sb[j + 1];
    const float q0 = qb[j], q1 = qb[j + 1];
    const float mx0 = (hasn > 0.f) ? fdec(mxb[j]) : 0.f;
    const float mx1 = (hasn > 0.f) ? fdec(mxb[j + 1]) : 0.f;
    const float mn0 = (hasn > 0.f) ? fdec(mnb[j]) : 0.f;
    const float mn1 = (hasn > 0.f) ? fdec(mnb[j + 1]) : 0.f;
    const float me0 = s0 * rdeg, me1 = s1 * rdeg;
    const float va0 = fmaxf(q0 * rdeg - me0 * me0, 0.f);
    const float va1 = fmaxf(q1 * rdeg - me1 * me1, 0.f);
    const float st0 = sqrtf(va0 + EPS_C), st1 = sqrtf(va1 + EPS_C);
    const float a6v0[6] = {me0, mx0, mn0, st0, va0, s0};
    const float a6v1[6] = {me1, mx1, mn1, st1, va1, s1};
#pragma unroll
    for (int c = 0; c < 6; ++c) {
      const int kk = c * FD + t * 4 + kh;
      v2f a  = {a6v0[c], a6v1[c]};
      v2f aa = {a.x * amp, a.y * amp};
      v2f at = {a.x * att, a.y * att};
#pragma unroll
      for (int nt = 0; nt < 4; ++nt) {
        const float* w1 = Wpow + (size_t)(64 + kk) * FD + nt * 16 + m;
        const float* w2 = Wpow + (size_t)(448 + kk) * FD + nt * 16 + m;
        const float* w3 = Wpow + (size_t)(832 + kk) * FD + nt * 16 + m;
        v2f b1 = {w1[0], w1[FD]};
        v2f b2 = {w2[0], w2[FD]};
        v2f b3 = {w3[0], w3[FD]};
        acc[nt] = wmma4(a, b1, acc[nt]);
        acc[nt] = wmma4(aa, b2, acc[nt]);
        acc[nt] = wmma4(at, b3, acc[nt]);
      }
    }
  }

  // stage post-MLP result (+bias) in LDS for the fused Wlin GEMM
  float* so = sOut[wv];
#pragma unroll
  for (int nt = 0; nt < 4; ++nt) {
    const int col = nt * 16 + m;
    const float bia = Wpob[col];
#pragma unroll
    for (int v = 0; v < 8; ++v) {
      const int row = v + ((lane < 16) ? 0 : 8);
      so[row * FD + col] = acc[nt][v] + bia;
    }
  }
  asm volatile("s_wait_asynccnt 0x0" ::: "memory");  // Wlin arrived in LDS
  __syncthreads();  // covers LDS stores of sOut + async fill visibility

  v8f acc2[4] = {vz, vz, vz, vz};
  for (int k0 = 0; k0 < FD; k0 += 4) {
    const int k = k0 + kh;
    v2f a = {so[m * FD + k], so[m * FD + k + 1]};
#pragma unroll
    for (int nt = 0; nt < 4; ++nt) {
      const float* wp = sWl + k * FD + nt * 16 + m;
      v2f b = {wp[0], wp[FD]};
      acc2[nt] = wmma4(a, b, acc2[nt]);
    }
  }

  // write out; in-wave BN partial reduction then one atomic per column
#pragma unroll
  for (int nt = 0; nt < 4; ++nt) {
    const int col = nt * 16 + m;
    const float bl = Wlb[col];
    float ps = 0.f, pq = 0.f;
#pragma unroll
    for (int v = 0; v < 8; ++v) {
      const int row = tile * 16 + v + ((lane < 16) ? 0 : 8);
      const float val = acc2[nt][v] + bl;
      out[(size_t)row * FD + col] = val;
      ps += val;
      pq += val * val;
    }
    ps += __shfl_xor(ps, 16, 32);  // lane and lane^16 hold the same column
    pq += __shfl_xor(pq, 16, 32);
    if (lane < 16) {
      atomicAdd(bnsum + col, ps);
      atomicAdd(bnsq + col, pq);
    }
  }
}

// ---------------------------------------------------------------- BN finalize
__global__ void k_bnfin(const float* __restrict__ bnsum,
                        const float* __restrict__ bnsq,
                        const float* __restrict__ g,
                        const float* __restrict__ b, float* __restrict__ ab) {
  int f = threadIdx.x;
  if (f < FD) {
    float mu  = bnsum[f] * (1.f / NN);
    float var = bnsq[f] * (1.f / NN) - mu * mu;
    float alpha = g[f] * rsqrtf(var + EPS_C);
    ab[f]      = alpha;
    ab[FD + f] = b[f] - mu * alpha;
  }
}

__global__ void k_final(const float* __restrict__ out,
                        const float* __restrict__ ab, float* __restrict__ x) {
  int i = blockIdx.x * blockDim.x + threadIdx.x;
  if (i >= NF_TOT) return;
  int col = i & (FD - 1);
  float o = out[i] * ab[col] + ab[FD + col];
  x[i] = fmaxf(o, 0.f) + x[i];
}

// ---------------------------------------------------------------- host
extern "C" void kernel_launch(void* const* d_in, const int* in_sizes, int n_in,
                              void* d_out, int out_size, void* d_ws,
                              size_t ws_size, hipStream_t stream) {
  const float* node_feat = (const float*)d_in[0];
  const int*   ei        = (const int*)d_in[1];
  const float* ef        = (const float*)d_in[2];
  const float* Wew       = (const float*)d_in[3];
  const float* Web       = (const float*)d_in[4];
  const float* Wpw       = (const float*)d_in[5];
  const float* Wpb       = (const float*)d_in[6];
  const float* Wpow      = (const float*)d_in[7];
  const float* Wpob      = (const float*)d_in[8];
  const float* Wlw       = (const float*)d_in[9];
  const float* Wlb       = (const float*)d_in[10];
  const float* bng       = (const float*)d_in[11];
  const float* bnb       = (const float*)d_in[12];

  float* ws = (float*)d_ws;
  float*    x     = ws;
  float*    out   = ws + (size_t)NF_TOT;
  float*    ssum  = ws + 2 * (size_t)NF_TOT;
  float*    ssq   = ws + 3 * (size_t)NF_TOT;
  unsigned* smax  = (unsigned*)(ws + 4 * (size_t)NF_TOT);
  unsigned* smin  = (unsigned*)(ws + 5 * (size_t)NF_TOT);
  float*    deg   = ws + 6 * (size_t)NF_TOT;
  float*    bnsum = deg + NN;
  float*    bnsq  = bnsum + FD;
  float*    ab    = bnsq + FD;

  const int tpb = 256;
  k_init<<<(NF_TOT + tpb - 1) / tpb, tpb, 0, stream>>>(node_feat, x, deg);
  k_deg<<<(NE + tpb - 1) / tpb, tpb, 0, stream>>>(ei, deg);

  for (int l = 0; l < NL; ++l) {
    k_reset<<<(NF_TOT + tpb - 1) / tpb, tpb, 0, stream>>>(ssum, ssq, smax,
                                                          smin, bnsum);
    k_edge<<<(NE / 16 + 7) / 8, tpb, 0, stream>>>(
        x, ei, ef, Wew + (size_t)l * DEF * FD, Web + (size_t)l * FD,
        Wpw + (size_t)l * 3 * FD * FD, Wpb + (size_t)l * FD, ssum, ssq, smax,
        smin);
    k_node<<<NN / 32, 64, 0, stream>>>(
        x, ssum, ssq, smax, smin, deg, Wpow + (size_t)l * 19 * FD * FD,
        Wpob + (size_t)l * FD, Wlw + (size_t)l * FD * FD, Wlb + (size_t)l * FD,
        out, bnsum, bnsq);
    k_bnfin<<<1, 64, 0, stream>>>(bnsum, bnsq, bng + (size_t)l * FD,
                                  bnb + (size_t)l * FD, ab);
    k_final<<<(NF_TOT + tpb - 1) / tpb, tpb, 0, stream>>>(out, ab, x);
  }

  (void)hipMemcpyAsync(d_out, x, (size_t)NF_TOT * sizeof(float),
                       hipMemcpyDeviceToDevice, stream);
}